// PointsRasterizer_3985729650843
// MI455X (gfx1250) — compile-verified
//
#include <hip/hip_runtime.h>

typedef float v2f __attribute__((ext_vector_type(2)));
typedef float v8f __attribute__((ext_vector_type(8)));
typedef unsigned int u32x4 __attribute__((ext_vector_type(4)));
typedef int i32x4 __attribute__((ext_vector_type(4)));
typedef int i32x8 __attribute__((ext_vector_type(8)));

#define Bq 2
#define Pq 4096
#define Hq 128
#define Wq 128
#define Kq 8
#define RADq   0.05f
#define RAD2q  (RADq * RADq)

// ---------------------------------------------------------------------------
// Prep: world -> view -> NDC, emit point features in WMMA A-matrix lane order.
//   A row (per point) = [-2x, -2y, x^2+y^2, 1]  (K=4)
//   a0 stream: tile t, lane l<16 -> f0[point t*16+l]; lane l>=16 -> f2
//   a1 stream: lane l<16 -> f1;                       lane l>=16 -> 1.0
//   z  stream: natural point order (depth key)
// Points behind the camera are poisoned (f0=f1=0, f2=1e30) so dist^2 >> R^2.
// ---------------------------------------------------------------------------
__global__ __launch_bounds__(256) void prep_kernel(
    const float* __restrict__ pts, const float* __restrict__ Rm,
    const float* __restrict__ Tm, const float* __restrict__ focal,
    float* __restrict__ ga0, float* __restrict__ ga1, float* __restrict__ gz)
{
    int tid = blockIdx.x * 256 + threadIdx.x;     // 0 .. B*P-1
    int b = tid >> 12;
    int p = tid & (Pq - 1);

    const float* pw = pts + tid * 3;
    float p0 = pw[0], p1 = pw[1], p2 = pw[2];
    const float* R = Rm + b * 9;
    const float* T = Tm + b * 3;
    float f = focal[b];

    // row-vector convention: X_view = X_world @ R + T
    float vx = p0 * R[0] + p1 * R[3] + p2 * R[6] + T[0];
    float vy = p0 * R[1] + p1 * R[4] + p2 * R[7] + T[1];
    float vz = p0 * R[2] + p1 * R[5] + p2 * R[8] + T[2];

    float inv = 1.0f / vz;
    float xn = f * vx * inv;
    float yn = f * vy * inv;

    float f0 = -2.0f * xn;
    float f1 = -2.0f * yn;
    float f2 = xn * xn + yn * yn;
    if (!(vz > 0.0f)) { f0 = 0.0f; f1 = 0.0f; f2 = 1e30f; }

    int ti = p >> 4, li = p & 15;
    int base = b * Pq + ti * 32 + li;
    ga0[base]      = f0;
    ga0[base + 16] = f2;
    ga1[base]      = f1;
    ga1[base + 16] = 1.0f;
    gz[b * Pq + p] = vz;
}

// ---------------------------------------------------------------------------
// TDM helper: 1-D tensor load of `nelem` f32 from global to LDS byte offset.
// D# bitfields per CDNA5 ISA:
//   group0: [1:0]count=1, [63:32]lds_addr, [120:64]global_addr, [127:126]type=2
//   group1: [17:16]data_size=2(4B), [79:48]tensor_dim0, [111:80]tensor_dim1=1,
//           [127:112]tile_dim0, [143:128]tile_dim1=1, [207:160]dim0_stride
// Tracked on TENSORcnt; wave issues regardless of EXEC.
// ---------------------------------------------------------------------------
#if __has_builtin(__builtin_amdgcn_tensor_load_to_lds)
#define HAVE_TDM 1
__device__ __forceinline__ void tdm_load_1d(const void* gsrc, unsigned lds_off,
                                            unsigned nelem)
{
    unsigned long long ga = (unsigned long long)(uintptr_t)gsrc;
    u32x4 g0;
    g0[0] = 1u;                                     // count=1 (valid user D#)
    g0[1] = lds_off;                                // LDS byte address
    g0[2] = (unsigned)ga;                           // global_addr[31:0]
    g0[3] = (unsigned)(ga >> 32) | (2u << 30);      // global_addr[56:32]|type=2
    i32x8 g1;
    g1[0] = (int)(2u << 16);                        // data_size = 4 bytes
    g1[1] = (int)((nelem & 0xffffu) << 16);         // tensor_dim0[15:0]
    g1[2] = (int)(((nelem >> 16) & 0xffffu) | (1u << 16)); // dim0 hi | dim1=1
    g1[3] = (int)((nelem & 0xffffu) << 16);         // tile_dim0
    g1[4] = 1;                                      // tile_dim1 = 1
    g1[5] = (int)nelem;                             // tensor_dim0_stride lo32
    g1[6] = 0;
    g1[7] = 0;
    i32x4 z4 = {0, 0, 0, 0};
#if defined(__clang_major__) && (__clang_major__ >= 23)
    i32x8 z8 = {0, 0, 0, 0, 0, 0, 0, 0};
    __builtin_amdgcn_tensor_load_to_lds(g0, g1, z4, z4, z8, 0);
#else
    __builtin_amdgcn_tensor_load_to_lds(g0, g1, z4, z4, 0);
#endif
}
#else
#define HAVE_TDM 0
#endif

// ---------------------------------------------------------------------------
// Raster: one block per image row (blockIdx = b*H + y), 8 waves * 16 pixels.
// Per wave: scan 256 point tiles; per surviving tile one v_wmma_f32_16x16x4_f32
// produces 16x16 exact squared distances; lane-local sorted top-8 by z; final
// cross-half merge via shfl_xor(16) + bitonic network.
// ---------------------------------------------------------------------------
__global__ __launch_bounds__(256) void raster_kernel(
    const float* __restrict__ ga0, const float* __restrict__ ga1,
    const float* __restrict__ gz, float* __restrict__ out)
{
    // Single LDS block at offset 0 so TDM lds_addr fields are constants.
    __shared__ __align__(16) float smem[3 * Pq];    // 48 KB
    float* sA0 = smem;
    float* sA1 = smem + Pq;
    float* sZ  = smem + 2 * Pq;

    const int tid = threadIdx.x;
    const int b   = blockIdx.x >> 7;        // H = 128
    const int y   = blockIdx.x & (Hq - 1);

#if HAVE_TDM
    // Stage 48 KB via Tensor Data Mover: wave 0 issues three 1-D descriptors,
    // waits TENSORcnt==0, then the workgroup barrier releases everyone.
    if (tid < 32) {
        tdm_load_1d(ga0 + b * Pq, 0u,                   Pq);
        tdm_load_1d(ga1 + b * Pq, (unsigned)(Pq * 4),   Pq);
        tdm_load_1d(gz  + b * Pq, (unsigned)(2 * Pq * 4), Pq);
        __builtin_amdgcn_s_wait_tensorcnt(0);
        // TDM names the LDS destination as an integer, so smem looks like a
        // never-stored, non-escaping global and its loads would fold to undef
        // (deleting the whole WMMA loop). This volatile read-modify-write is a
        // real, un-removable store to smem (writes back the TDM-loaded value,
        // so it is a semantic no-op) that forces the ds_loads to stay.
        if (tid == 0) {
            volatile float* guard = smem;
            float g = guard[0];
            guard[0] = g;
        }
    }
    asm volatile("" ::: "memory");
#else
    {
        const float4* g0 = (const float4*)(ga0 + b * Pq);
        const float4* g1 = (const float4*)(ga1 + b * Pq);
        const float4* g2 = (const float4*)(gz  + b * Pq);
        float4* s0 = (float4*)sA0;
        float4* s1 = (float4*)sA1;
        float4* s2 = (float4*)sZ;
#pragma unroll
        for (int i = 0; i < 4; ++i) {
            int j = tid + i * 256;
            s0[j] = g0[j];
            s1[j] = g1[j];
            s2[j] = g2[j];
        }
    }
#endif
    __syncthreads();

    const int lane = tid & 31;
    const int wave = tid >> 5;
    const int half = lane >> 4;
    const int li   = lane & 15;
    const int x0   = wave * 16;
    const int px   = x0 + li;

    const float gy = 1.0f - 2.0f * ((float)y  + 0.5f) * (1.0f / Hq);
    const float gx = 1.0f - 2.0f * ((float)px + 0.5f) * (1.0f / Wq);

    // B matrix (4x16): rows [gx, gy, 1, gx^2+gy^2] striped across lane halves.
    v2f bm;
    bm[0] = half ? gy : gx;
    bm[1] = half ? (gx * gx + gy * gy) : 1.0f;

    // Tile cull bounds: gx is decreasing in pixel x.
    const float gxmin = 1.0f - 2.0f * ((float)x0 + 15.5f) * (1.0f / Wq);
    const float gxmax = 1.0f - 2.0f * ((float)x0 + 0.5f)  * (1.0f / Wq);
    const float RM = RADq + 1e-3f;

    // Lane-local sorted-ascending top-8 (z key) with payload (idx, dist2).
    float zs[Kq]; int is[Kq]; float dr[Kq];
#pragma unroll
    for (int k = 0; k < Kq; ++k) { zs[k] = __builtin_inff(); is[k] = -1; dr[k] = -1.0f; }

    // Software-pipelined tile scan: ds_load latency of tile t+1 overlaps the
    // cull / WMMA / top-K work of tile t.
    float a0 = sA0[lane];
    float a1 = sA1[lane];
    for (int t = 0; t < Pq / 16; ++t) {
        int tn = (t + 1) & (Pq / 16 - 1);
        float na0 = sA0[tn * 32 + lane];
        float na1 = sA1[tn * 32 + lane];

        // Bbox cull: lanes 0-15 hold (-2x, -2y) of the 16 tile points.
        float xpt = -0.5f * a0;
        float ypt = -0.5f * a1;
        bool near = (half == 0) & (xpt > gxmin - RM) & (xpt < gxmax + RM) &
                    (__builtin_fabsf(ypt - gy) < RM);
        if (__builtin_amdgcn_ballot_w32(near)) {   // uniform: EXEC stays all-1s
            v2f am; am[0] = a0; am[1] = a1;
            v8f c = {0.f, 0.f, 0.f, 0.f, 0.f, 0.f, 0.f, 0.f};
            // D[m][n] = dist^2(point m, pixel n), exact rank-4 factorization.
            v8f d = __builtin_amdgcn_wmma_f32_16x16x4_f32(
                false, am, false, bm, (short)0, c, false, false);

            const float4* zp = (const float4*)(sZ + t * 16 + half * 8);
            float4 zA = zp[0];
            float4 zB = zp[1];
            float zl[8] = {zA.x, zA.y, zA.z, zA.w, zB.x, zB.y, zB.z, zB.w};
            int pbase = t * 16 + half * 8;

#pragma unroll
            for (int r = 0; r < 8; ++r) {
                float dd = d[r];
                if (dd < RAD2q) {                  // rare: exec-masked insert
                    float zz = zl[r];
                    int   ii = pbase + r;
                    bool lt[Kq];
#pragma unroll
                    for (int j = 0; j < Kq; ++j) lt[j] = zz < zs[j];
#pragma unroll
                    for (int j = Kq - 1; j >= 1; --j) {
                        zs[j] = lt[j-1] ? zs[j-1] : (lt[j] ? zz : zs[j]);
                        is[j] = lt[j-1] ? is[j-1] : (lt[j] ? ii : is[j]);
                        dr[j] = lt[j-1] ? dr[j-1] : (lt[j] ? dd : dr[j]);
                    }
                    if (lt[0]) { zs[0] = zz; is[0] = ii; dr[0] = dd; }
                }
            }
        }
        a0 = na0;
        a1 = na1;
    }

    // Merge the two lane-half top-8 lists for each pixel.
    float rz[Kq]; int ri[Kq]; float rd[Kq];
#pragma unroll
    for (int k = 0; k < Kq; ++k) {
        rz[k] = __shfl_xor(zs[k], 16, 32);
        ri[k] = __shfl_xor(is[k], 16, 32);
        rd[k] = __shfl_xor(dr[k], 16, 32);
    }
    // Bitonic first stage: lower half of the 16-element merge (result bitonic).
    float mz[Kq]; int mi[Kq]; float md[Kq];
#pragma unroll
    for (int k = 0; k < Kq; ++k) {
        float za = zs[k], zb2 = rz[7 - k];
        bool ta = (za < zb2) || ((za == zb2) && (is[k] < ri[7 - k]));
        mz[k] = ta ? za : zb2;
        mi[k] = ta ? is[k] : ri[7 - k];
        md[k] = ta ? dr[k] : rd[7 - k];
    }
    // Sort the bitonic 8 with a 3-stage bitonic merge network.
#define CEX(i, j)                                                              \
    {                                                                          \
        bool sw = (mz[i] > mz[j]) || ((mz[i] == mz[j]) && (mi[i] > mi[j]));    \
        float tz = mz[i]; int tii = mi[i]; float td = md[i];                   \
        mz[i] = sw ? mz[j] : mz[i]; mi[i] = sw ? mi[j] : mi[i];                \
        md[i] = sw ? md[j] : md[i];                                            \
        mz[j] = sw ? tz : mz[j]; mi[j] = sw ? tii : mi[j];                     \
        md[j] = sw ? td : md[j];                                               \
    }
    CEX(0, 4) CEX(1, 5) CEX(2, 6) CEX(3, 7)
    CEX(0, 2) CEX(1, 3) CEX(4, 6) CEX(5, 7)
    CEX(0, 1) CEX(2, 3) CEX(4, 5) CEX(6, 7)
#undef CEX

    if (lane < 16) {
        const int BHWK = Bq * Hq * Wq * Kq;          // 262144
        int pix = blockIdx.x * Wq + px;              // (b*H + y)*W + x
        float oi[Kq], oz[Kq], od[Kq];
#pragma unroll
        for (int k = 0; k < Kq; ++k) {
            bool val = mz[k] < 3.0e38f;
            oi[k] = val ? (float)mi[k] : -1.0f;
            oz[k] = val ? mz[k]        : -1.0f;
            od[k] = val ? md[k]        : -1.0f;
        }
        float4* pI = (float4*)(out + pix * Kq);
        float4* pZ = (float4*)(out + BHWK + pix * Kq);
        float4* pD = (float4*)(out + 2 * BHWK + pix * Kq);
        pI[0] = make_float4(oi[0], oi[1], oi[2], oi[3]);
        pI[1] = make_float4(oi[4], oi[5], oi[6], oi[7]);
        pZ[0] = make_float4(oz[0], oz[1], oz[2], oz[3]);
        pZ[1] = make_float4(oz[4], oz[5], oz[6], oz[7]);
        pD[0] = make_float4(od[0], od[1], od[2], od[3]);
        pD[1] = make_float4(od[4], od[5], od[6], od[7]);
    }
}

extern "C" void kernel_launch(void* const* d_in, const int* in_sizes, int n_in,
                              void* d_out, int out_size, void* d_ws, size_t ws_size,
                              hipStream_t stream) {
    const float* pts   = (const float*)d_in[0];
    const float* R     = (const float*)d_in[1];
    const float* T     = (const float*)d_in[2];
    const float* focal = (const float*)d_in[3];

    float* wsf = (float*)d_ws;
    float* ga0 = wsf;                    // B*P floats, WMMA-A k0/k2 stream
    float* ga1 = wsf + Bq * Pq;          // B*P floats, WMMA-A k1/k3 stream
    float* gz  = wsf + 2 * Bq * Pq;      // B*P floats, view-space z

    prep_kernel<<<(Bq * Pq) / 256, 256, 0, stream>>>(pts, R, T, focal, ga0, ga1, gz);
    raster_kernel<<<Bq * Hq, 256, 0, stream>>>(ga0, ga1, gz, (float*)d_out);
}